// Model_43645457662087
// MI455X (gfx1250) — compile-verified
//
#include <hip/hip_runtime.h>
#include <math.h>

typedef __attribute__((ext_vector_type(2))) float v2f;
typedef __attribute__((ext_vector_type(8))) float v8f;

#define BB 16
#define LL 1024
#define HH 8
#define EE 64
#define TT 64
#define OO 64
#define SS (BB*HH)      /* 128 slices (b,h) */
#define FF 513
#define FPAD 528
#define SEL (SS*TT*EE)  /* 524288 */

#define CH 128          /* l-rows per LDS chunk  */
#define RP 72           /* padded LDS row stride (floats): half0/half1 hit disjoint banks */

static __device__ __forceinline__ v8f wmma4(v2f a, v2f b, v8f c) {
  // V_WMMA_F32_16X16X4_F32 : D = A(16x4) * B(4x16) + C(16x16), all f32
  return __builtin_amdgcn_wmma_f32_16x16x4_f32(false, a, false, b, (short)0, c, false, false);
}

static __device__ __forceinline__ v8f vzero8() {
  v8f z = {0.f,0.f,0.f,0.f,0.f,0.f,0.f,0.f};
  return z;
}

// gfx1250 async DMA: global -> LDS, 16B per lane, tracked by ASYNCcnt.
static __device__ __forceinline__ void async_b128(unsigned int lds_off, const void* g) {
  asm volatile("global_load_async_to_lds_b128 %0, %1, off"
               :: "v"(lds_off), "v"(g) : "memory");
}
static __device__ __forceinline__ void wait_async0() {
  asm volatile("s_wait_asynccnt 0x0" ::: "memory");
}

// ---------------------------------------------------------------------------
// Kernel 1: twiddle table tw[f][l] = (cos(2*pi*f*l/L), -sin(2*pi*f*l/L))
// rows [FF, FPAD) zero-padded so WMMA f-tiles can overrun harmlessly.
// ---------------------------------------------------------------------------
__global__ void k_twiddle(float2* tw) {
  int i = blockIdx.x * blockDim.x + threadIdx.x;
  if (i >= FPAD * LL) return;
  int f = i >> 10, l = i & 1023;
  float2 r;
  if (f < FF) {
    int m = (f * l) & 1023;
    float ang = -6.28318530717958647692f * (float)m * (1.0f / 1024.0f);
    r.x = cosf(ang);
    r.y = sinf(ang);   // == -sin(2*pi*f*l/L): rfft imaginary convention baked in
  } else {
    r.x = 0.f; r.y = 0.f;
  }
  tw[i] = r;
}

// ---------------------------------------------------------------------------
// Kernel 2: zero mag accumulator; per-indexq output scale.
// cq[j] = last-dup-wins * (f==0||f==L/2 ? 1 : 2) / (L*IN_CH*OUT_CH)   (=2^-28)
// ---------------------------------------------------------------------------
__global__ void k_prep(const int* __restrict__ indexq, float* mag, float* cqv) {
  int t = threadIdx.x;
  if (t < FPAD) mag[t] = 0.0f;
  if (t < TT) {
    int f = indexq[t];
    int valid = 1;
    for (int j = t + 1; j < TT; ++j)
      if (indexq[j] == f) valid = 0;   // reference scatter .set -> last write wins
    float c = (f == 0 || f == (LL / 2)) ? 1.0f : 2.0f;
    cqv[t] = valid ? c * (1.0f / 268435456.0f) : 0.0f;
  }
}

// ---------------------------------------------------------------------------
// Kernel 3: mag[f] += sum_e |DFT_f(k_slice[:,e])| over all 128 slices.
// A = twiddles (16f x K), B = k slice staged via async DMA into LDS,
// double-buffered 32KB chunks so the DMA overlaps the WMMA stream.
// ---------------------------------------------------------------------------
__global__ void __launch_bounds__(128) k_kmag(const float* __restrict__ kin,
                                              const float2* __restrict__ tw,
                                              float* mag) {
  __shared__ float lbuf[2][CH * RP];
  int s = blockIdx.x;
  int bb = s >> 3, h = s & 7;
  int tid = threadIdx.x;
  int w = tid >> 5, lane = tid & 31, half = lane >> 4, lp = lane & 15;
  const float* kslice = kin + (size_t)bb * (LL * HH * EE) + (size_t)h * EE;
  int l0s = tid >> 4;          // 0..7 : staging row within group of 8
  int e4  = tid & 15;          // 0..15: 16B column within row
  unsigned int lb0 = (unsigned int)(uintptr_t)&lbuf[0][0];
  unsigned int lb1 = (unsigned int)(uintptr_t)&lbuf[1][0];

  for (int ftg = 0; ftg < 9; ++ftg) {
    int ft = ftg * 4 + w;
    bool act = (ft < 33);
    int f0 = act ? ft * 16 : 0;
    const float4* twrow4 = (const float4*)tw + (size_t)(f0 + lp) * (LL / 2);
    v8f ar[4], ai[4];
    for (int n = 0; n < 4; ++n) { ar[n] = vzero8(); ai[n] = vzero8(); }

    // stage chunk 0
    for (int i = 0; i < 16; ++i) {
      int l = l0s + i * 8;
      async_b128(lb0 + (unsigned int)(l * RP * 4 + e4 * 16),
                 kslice + (size_t)l * (HH * EE) + e4 * 4);
    }
    wait_async0();
    __syncthreads();

    for (int c = 0; c < 8; ++c) {
      int buf = c & 1;
      if (c + 1 < 8) {                       // prefetch next chunk into other buffer
        const float* gp = kslice + (size_t)(c + 1) * CH * (HH * EE);
        unsigned int lb = ((c + 1) & 1) ? lb1 : lb0;
        for (int i = 0; i < 16; ++i) {
          int l = l0s + i * 8;
          async_b128(lb + (unsigned int)(l * RP * 4 + e4 * 16),
                     gp + (size_t)l * (HH * EE) + e4 * 4);
        }
      }
      if (act) {
        const float* lb = lbuf[buf];
        for (int kk = 0; kk < 32; ++kk) {
          int lrel = kk * 4 + 2 * half;
          float4 t4 = twrow4[(size_t)c * 64 + kk * 2 + half];
          v2f ac = { t4.x, t4.z };
          v2f as = { t4.y, t4.w };
          const float* r0 = lb + lrel * RP;
          for (int et = 0; et < 4; ++et) {
            int e = et * 16 + lp;
            v2f bv = { r0[e], r0[RP + e] };
            ar[et] = wmma4(ac, bv, ar[et]);
            ai[et] = wmma4(as, bv, ai[et]);
          }
        }
      }
      wait_async0();
      __syncthreads();
    }

    if (act) {
      for (int j = 0; j < 8; ++j) {
        float sj = 0.f;
        for (int et = 0; et < 4; ++et) {
          float re = ar[et][j], im = ai[et][j];
          sj += sqrtf(re * re + im * im);
        }
        sj += __shfl_xor(sj, 1, 16);
        sj += __shfl_xor(sj, 2, 16);
        sj += __shfl_xor(sj, 4, 16);
        sj += __shfl_xor(sj, 8, 16);
        int f = f0 + j + 8 * half;
        if (lp == 0 && f < FF) atomicAdd(&mag[f], sj);
      }
    }
  }
}

// ---------------------------------------------------------------------------
// Kernel 4: top-64 of mag[1..512] (bin 0 forced out, ties -> lower index).
// ---------------------------------------------------------------------------
__global__ void __launch_bounds__(1024) k_topk(const float* __restrict__ mag,
                                               int* kidx) {
  __shared__ float vk[FPAD];
  __shared__ float v[1024];
  __shared__ int   id[1024];
  int t = threadIdx.x;
  if (t < FPAD) vk[t] = (t > 0 && t < FF) ? mag[t] : -1.0f;
  __syncthreads();
  for (int it = 0; it < TT; ++it) {
    v[t] = (t < FF) ? vk[t] : -2.0f;
    id[t] = t;
    __syncthreads();
    for (int strd = 512; strd > 0; strd >>= 1) {
      if (t < strd) {
        float vo = v[t + strd]; int io = id[t + strd];
        if (vo > v[t] || (vo == v[t] && io < id[t])) { v[t] = vo; id[t] = io; }
      }
      __syncthreads();
    }
    if (t == 0) { kidx[it] = id[0]; vk[id[0]] = -1.0f; }
    __syncthreads();
  }
}

// ---------------------------------------------------------------------------
// Kernel 5: gathered DFT. src=0: q @ indexq -> (qre,qim); src=1: k @ kidx.
// Source slice staged through LDS (async, double-buffered). Output [s][freq][e].
// ---------------------------------------------------------------------------
__global__ void __launch_bounds__(128) k_gdft(const float* __restrict__ qin,
                                              const float* __restrict__ kin,
                                              const int* __restrict__ indexq,
                                              const int* __restrict__ kidx,
                                              const float2* __restrict__ tw,
                                              float* qre, float* qim,
                                              float* kre, float* kim) {
  __shared__ float lbuf[2][CH * RP];
  int src = blockIdx.x & 1;
  int s   = blockIdx.x >> 1;
  int bb = s >> 3, h = s & 7;
  const float* din = src ? kin : qin;
  float* dre = src ? kre : qre;
  float* dim = src ? kim : qim;
  int tid = threadIdx.x;
  int w = tid >> 5, lane = tid & 31, half = lane >> 4, lp = lane & 15;
  const float* dslice = din + (size_t)bb * (LL * HH * EE) + (size_t)h * EE;
  int l0s = tid >> 4, e4 = tid & 15;
  unsigned int lb0 = (unsigned int)(uintptr_t)&lbuf[0][0];
  unsigned int lb1 = (unsigned int)(uintptr_t)&lbuf[1][0];

  int m = w * 16 + lp;
  int f = src ? kidx[m] : indexq[m];
  const float4* twrow4 = (const float4*)tw + (size_t)f * (LL / 2);

  v8f ar[4], ai[4];
  for (int n = 0; n < 4; ++n) { ar[n] = vzero8(); ai[n] = vzero8(); }

  for (int i = 0; i < 16; ++i) {
    int l = l0s + i * 8;
    async_b128(lb0 + (unsigned int)(l * RP * 4 + e4 * 16),
               dslice + (size_t)l * (HH * EE) + e4 * 4);
  }
  wait_async0();
  __syncthreads();

  for (int c = 0; c < 8; ++c) {
    int buf = c & 1;
    if (c + 1 < 8) {
      const float* gp = dslice + (size_t)(c + 1) * CH * (HH * EE);
      unsigned int lb = ((c + 1) & 1) ? lb1 : lb0;
      for (int i = 0; i < 16; ++i) {
        int l = l0s + i * 8;
        async_b128(lb + (unsigned int)(l * RP * 4 + e4 * 16),
                   gp + (size_t)l * (HH * EE) + e4 * 4);
      }
    }
    {
      const float* lb = lbuf[buf];
      for (int kk = 0; kk < 32; ++kk) {
        int lrel = kk * 4 + 2 * half;
        float4 t4 = twrow4[(size_t)c * 64 + kk * 2 + half];
        v2f ac = { t4.x, t4.z };
        v2f as = { t4.y, t4.w };
        const float* r0 = lb + lrel * RP;
        for (int et = 0; et < 4; ++et) {
          int e = et * 16 + lp;
          v2f bv = { r0[e], r0[RP + e] };
          ar[et] = wmma4(ac, bv, ar[et]);
          ai[et] = wmma4(as, bv, ai[et]);
        }
      }
    }
    wait_async0();
    __syncthreads();
  }

  for (int et = 0; et < 4; ++et)
    for (int j = 0; j < 8; ++j) {
      int row = w * 16 + j + 8 * half;
      int col = et * 16 + lp;
      size_t o = ((size_t)s * TT + row) * EE + col;
      dre[o] = ar[et][j];
      dim[o] = ai[et][j];
    }
}

// ---------------------------------------------------------------------------
// Kernel 6 (per slice): xqk = Xq . Xk^T (contract e) -> complex tanh (LDS)
//                       xqkv[e][x] = sum_y tanh(xqk)[x][y] * Xk[y][e]
// ---------------------------------------------------------------------------
__global__ void __launch_bounds__(128) k_attn(const float* __restrict__ qre,
                                              const float* __restrict__ qim,
                                              const float* __restrict__ kre,
                                              const float* __restrict__ kim,
                                              float* vre, float* vim) {
  __shared__ float tre[TT * TT];
  __shared__ float tim[TT * TT];
  int s = blockIdx.x;
  int tid = threadIdx.x;
  int w = tid >> 5, lane = tid & 31, half = lane >> 4, lp = lane & 15;
  size_t sb = (size_t)s * (TT * EE);

  { // phase 1: xqk + tanh
    int mt = w;
    int x = mt * 16 + lp;
    v8f rr[4], ii[4];
    for (int n = 0; n < 4; ++n) { rr[n] = vzero8(); ii[n] = vzero8(); }
    for (int kk = 0; kk < 16; ++kk) {
      int e0 = kk * 4 + 2 * half;
      v2f aqr = { qre[sb + (size_t)x * EE + e0], qre[sb + (size_t)x * EE + e0 + 1] };
      v2f aqi = { qim[sb + (size_t)x * EE + e0], qim[sb + (size_t)x * EE + e0 + 1] };
      v2f aqn = { -aqi.x, -aqi.y };
      for (int nt = 0; nt < 4; ++nt) {
        int y = nt * 16 + lp;
        v2f bkr = { kre[sb + (size_t)y * EE + e0], kre[sb + (size_t)y * EE + e0 + 1] };
        v2f bki = { kim[sb + (size_t)y * EE + e0], kim[sb + (size_t)y * EE + e0 + 1] };
        rr[nt] = wmma4(aqr, bkr, rr[nt]);
        rr[nt] = wmma4(aqn, bki, rr[nt]);   // re -= qi*ki
        ii[nt] = wmma4(aqr, bki, ii[nt]);
        ii[nt] = wmma4(aqi, bkr, ii[nt]);
      }
    }
    for (int nt = 0; nt < 4; ++nt)
      for (int j = 0; j < 8; ++j) {
        int xr = mt * 16 + j + 8 * half;
        int y  = nt * 16 + lp;
        float a = rr[nt][j], b = ii[nt][j];
        float tr, ti;
        if (fabsf(a) > 12.0f) {             // tanh saturates, imag ~ e^{-2|a|}
          tr = copysignf(1.0f, a); ti = 0.0f;
        } else {
          float d = coshf(2.0f * a) + cosf(2.0f * b);
          tr = sinhf(2.0f * a) / d;
          ti = sinf(2.0f * b) / d;
        }
        tre[xr * TT + y] = tr;
        tim[xr * TT + y] = ti;
      }
  }
  __syncthreads();
  { // phase 2: xqkv (contract y)
    int et = w;
    int e = et * 16 + lp;
    v8f rr[4], ii[4];
    for (int n = 0; n < 4; ++n) { rr[n] = vzero8(); ii[n] = vzero8(); }
    for (int kk = 0; kk < 16; ++kk) {
      int y0 = kk * 4 + 2 * half;
      v2f akr = { kre[sb + (size_t)y0 * EE + e], kre[sb + (size_t)(y0 + 1) * EE + e] };
      v2f aki = { kim[sb + (size_t)y0 * EE + e], kim[sb + (size_t)(y0 + 1) * EE + e] };
      v2f akn = { -aki.x, -aki.y };
      for (int nt = 0; nt < 4; ++nt) {
        int x = nt * 16 + lp;
        v2f btr = { tre[x * TT + y0], tre[x * TT + y0 + 1] };
        v2f bti = { tim[x * TT + y0], tim[x * TT + y0 + 1] };
        rr[nt] = wmma4(akr, btr, rr[nt]);
        rr[nt] = wmma4(akn, bti, rr[nt]);   // re -= ki*ti
        ii[nt] = wmma4(akr, bti, ii[nt]);
        ii[nt] = wmma4(aki, btr, ii[nt]);
      }
    }
    for (int nt = 0; nt < 4; ++nt)
      for (int j = 0; j < 8; ++j) {
        int er = et * 16 + j + 8 * half;
        int x  = nt * 16 + lp;
        size_t o = sb + (size_t)er * TT + x;
        vre[o] = rr[nt][j];
        vim[o] = ii[nt][j];
      }
  }
}

// ---------------------------------------------------------------------------
// Kernel 7: per (h,x): out[b][o] = sum_e xqkv[b,h,e,x] * w[h,e,o,x] (complex)
// ---------------------------------------------------------------------------
__global__ void __launch_bounds__(64) k_wproj(const float* __restrict__ vre,
                                              const float* __restrict__ vim,
                                              const float* __restrict__ wr,
                                              const float* __restrict__ wi,
                                              float* owre, float* owim) {
  int h = blockIdx.x >> 6;
  int x = blockIdx.x & 63;
  int tid = threadIdx.x;
  int w = tid >> 5, lane = tid & 31, half = lane >> 4, lp = lane & 15;
  v8f rr[2], ii[2];
  for (int n = 0; n < 2; ++n) { rr[n] = vzero8(); ii[n] = vzero8(); }
  size_t abase = ((size_t)lp * HH + h) * (EE * TT) + x;   // + e*TT
  size_t wbase = (size_t)h * (EE * OO * TT) + x;          // + e*OO*TT + o*TT
  for (int kk = 0; kk < 16; ++kk) {
    int e0 = kk * 4 + 2 * half;
    v2f ar_ = { vre[abase + (size_t)e0 * TT], vre[abase + (size_t)(e0 + 1) * TT] };
    v2f ai_ = { vim[abase + (size_t)e0 * TT], vim[abase + (size_t)(e0 + 1) * TT] };
    v2f an_ = { -ai_.x, -ai_.y };
    for (int t = 0; t < 2; ++t) {
      int nt = w + 2 * t;
      int o = nt * 16 + lp;
      size_t wb = wbase + (size_t)e0 * (OO * TT) + (size_t)o * TT;
      v2f br = { wr[wb], wr[wb + OO * TT] };
      v2f bi = { wi[wb], wi[wb + OO * TT] };
      rr[t] = wmma4(ar_, br, rr[t]);
      rr[t] = wmma4(an_, bi, rr[t]);
      ii[t] = wmma4(ar_, bi, ii[t]);
      ii[t] = wmma4(ai_, br, ii[t]);
    }
  }
  for (int t = 0; t < 2; ++t) {
    int nt = w + 2 * t;
    for (int j = 0; j < 8; ++j) {
      int b = j + 8 * half;
      int o = nt * 16 + lp;
      size_t oo = (((size_t)b * HH + h) * OO + o) * TT + x;
      owre[oo] = rr[t][j];
      owim[oo] = ii[t][j];
    }
  }
}

// ---------------------------------------------------------------------------
// Kernel 8: sparse iRFFT: out[s][o][l] = sum_j cq_j*(Re[o][j]*tw.x + Im[o][j]*tw.y)
// (tw.y already holds -sin, so both terms are additive.)
// ---------------------------------------------------------------------------
__global__ void __launch_bounds__(256) k_irfft(const float* __restrict__ owre,
                                               const float* __restrict__ owim,
                                               const int* __restrict__ indexq,
                                               const float* __restrict__ cqv,
                                               const float2* __restrict__ tw,
                                               float* __restrict__ out) {
  int s = blockIdx.x;
  int tid = threadIdx.x;
  int w = tid >> 5, lane = tid & 31, half = lane >> 4, lp = lane & 15;
  size_t sb = (size_t)s * (OO * TT);
  size_t ob = (size_t)s * (OO * LL);
  for (int lt = w; lt < 64; lt += 8) {
    int l0 = lt * 16;
    v8f acc[4];
    for (int n = 0; n < 4; ++n) acc[n] = vzero8();
    for (int kk = 0; kk < 16; ++kk) {
      int j0 = kk * 4 + 2 * half;
      int f0 = indexq[j0], f1 = indexq[j0 + 1];
      float c0 = cqv[j0], c1 = cqv[j0 + 1];
      float2 t0 = tw[(size_t)f0 * LL + l0 + lp];
      float2 t1 = tw[(size_t)f1 * LL + l0 + lp];
      v2f bc = { c0 * t0.x, c1 * t1.x };
      v2f bs = { c0 * t0.y, c1 * t1.y };
      for (int ot = 0; ot < 4; ++ot) {
        int o = ot * 16 + lp;
        v2f ar_ = { owre[sb + (size_t)o * TT + j0], owre[sb + (size_t)o * TT + j0 + 1] };
        v2f ai_ = { owim[sb + (size_t)o * TT + j0], owim[sb + (size_t)o * TT + j0 + 1] };
        acc[ot] = wmma4(ar_, bc, acc[ot]);
        acc[ot] = wmma4(ai_, bs, acc[ot]);
      }
    }
    for (int ot = 0; ot < 4; ++ot)
      for (int j = 0; j < 8; ++j) {
        int o = ot * 16 + j + 8 * half;
        out[ob + (size_t)o * LL + l0 + lp] = acc[ot][j];
      }
  }
}

// ---------------------------------------------------------------------------
extern "C" void kernel_launch(void* const* d_in, const int* in_sizes, int n_in,
                              void* d_out, int out_size, void* d_ws, size_t ws_size,
                              hipStream_t stream) {
  const float* q      = (const float*)d_in[0];
  const float* k      = (const float*)d_in[1];
  /* d_in[2] = v      : unused by the reference */
  const int*   indexq = (const int*)d_in[3];
  /* d_in[4] = indexk : unused by the reference */
  const float* wr     = (const float*)d_in[5];
  const float* wi     = (const float*)d_in[6];
  float* out = (float*)d_out;

  float*  ws  = (float*)d_ws;
  float2* tw  = (float2*)ws;                       // FPAD*LL float2
  float*  mag = ws + 2 * (size_t)FPAD * LL;        // 528
  float*  cqv = mag + FPAD;                        // 64
  int*    kidx = (int*)(cqv + TT);                 // 64
  float*  qre = (float*)(kidx + TT);
  float*  qim = qre + SEL;
  float*  kre = qim + SEL;
  float*  kim = kre + SEL;
  float*  vre = kim + SEL;
  float*  vim = vre + SEL;
  float*  ore = vim + SEL;
  float*  oim = ore + SEL;

  (void)in_sizes; (void)n_in; (void)out_size; (void)ws_size;

  k_twiddle<<<(FPAD * LL + 255) / 256, 256, 0, stream>>>(tw);
  k_prep   <<<1, 576, 0, stream>>>(indexq, mag, cqv);
  k_kmag   <<<SS, 128, 0, stream>>>(k, tw, mag);
  k_topk   <<<1, 1024, 0, stream>>>(mag, kidx);
  k_gdft   <<<SS * 2, 128, 0, stream>>>(q, k, indexq, kidx, tw, qre, qim, kre, kim);
  k_attn   <<<SS, 128, 0, stream>>>(qre, qim, kre, kim, vre, vim);
  k_wproj  <<<HH * TT, 64, 0, stream>>>(vre, vim, wr, wi, ore, oim);
  k_irfft  <<<SS, 256, 0, stream>>>(ore, oim, indexq, cqv, tw, out);
}